// NemotronHMamba2_91164975825164
// MI455X (gfx1250) — compile-verified
//
#include <hip/hip_runtime.h>
#include <hip/hip_bf16.h>

// ---------------------------------------------------------------------------
// Mamba2 (NemotronH) forward for gfx1250 (MI455X).
// fp32 WMMA GEMMs with double-buffered GLOBAL_LOAD_ASYNC_TO_LDS_B128 staging.
// Shapes: B=1, L=512, HID=4096, H=64, P=64, N=128, G=8, K=4
// ---------------------------------------------------------------------------

#define L_     512
#define HID_   4096
#define H_     64
#define P_     64
#define N_     128
#define G_     8
#define KCONV  4
#define INTER_ 4096
#define GN_    1024
#define CONVD  6144
#define PROJ_  10304
#define GS_    512
#define EPS_   1e-5f

typedef float v2f __attribute__((ext_vector_type(2)));
typedef float v8f __attribute__((ext_vector_type(8)));

#define LDS_STRIDE 36   // 32 floats + 4 pad: 144B rows -> 16B aligned, bank-spread

// One 16-byte async DMA transfer: global -> LDS, tracked by ASYNCcnt.
__device__ __forceinline__ void async_copy16(unsigned lds_off, const float* gptr)
{
    asm volatile("global_load_async_to_lds_b128 %0, %1, off"
                 :: "v"(lds_off), "v"((unsigned long long)(uintptr_t)gptr)
                 : "memory");
}

// ---------------------------------------------------------------------------
// GEMM: C[M,N] = A[M,K] * W[N,K]^T  using V_WMMA_F32_16X16X4_F32.
// Block = 256 threads (8 waves), computes a 128x64 tile of C.
// K is consumed in 32-wide tiles staged in LDS (double buffered, async DMA).
// grid.x = N/64, grid.y = M/128. Requires M%128==0, N%64==0, K%32==0.
// ---------------------------------------------------------------------------
__global__ __launch_bounds__(256) void wmma_gemm_xWt(
    const float* __restrict__ A, const float* __restrict__ W,
    float* __restrict__ C, int M, int N, int K)
{
    __shared__ __align__(16) float sA[2][128 * LDS_STRIDE];
    __shared__ __align__(16) float sW[2][64 * LDS_STRIDE];

    const int tid  = threadIdx.x;
    const int lane = tid & 31;
    const int wave = tid >> 5;
    const int row0 = blockIdx.y * 128;
    const int col0 = blockIdx.x * 64;

    // --- async-copy lane mapping: 8 lanes x 16B cover one 32-float row ---
    const int arow = tid >> 3;          // 0..31 (per 32-row issue)
    const int acol = (tid & 7) * 4;     // float column within K-tile
    const float* gA = A + (size_t)(row0 + arow) * K + acol;
    const float* gW = W + (size_t)(col0 + arow) * K + acol;
    const unsigned aoff   = (unsigned)((arow * LDS_STRIDE + acol) * 4);
    const unsigned baseA0 = (unsigned)(uintptr_t)&sA[0][0];
    const unsigned baseA1 = (unsigned)(uintptr_t)&sA[1][0];
    const unsigned baseW0 = (unsigned)(uintptr_t)&sW[0][0];
    const unsigned baseW1 = (unsigned)(uintptr_t)&sW[1][0];

    // 6 async issues per thread per K-tile (4 for A's 128 rows, 2 for W's 64)
    auto issue_tile = [&](int kt, int buf) {
        const float* ga = gA + (size_t)kt * 32;
        const float* gw = gW + (size_t)kt * 32;
        const unsigned la = (buf ? baseA1 : baseA0) + aoff;
        const unsigned lw = (buf ? baseW1 : baseW0) + aoff;
#pragma unroll
        for (int i = 0; i < 4; ++i)
            async_copy16(la + (unsigned)(i * 32 * LDS_STRIDE * 4),
                         ga + (size_t)i * 32 * K);
#pragma unroll
        for (int i = 0; i < 2; ++i)
            async_copy16(lw + (unsigned)(i * 32 * LDS_STRIDE * 4),
                         gw + (size_t)i * 32 * K);
    };

    // --- WMMA fragment mapping (f32 16x16x4, ISA 7.12.2) ---
    const int frow  = (lane & 15);
    const int khalf = (lane >> 4) << 1;          // 0 or 2
    const int mloc  = wave * 16 + frow;          // A row within block tile

    v8f acc0 = {}, acc1 = {}, acc2 = {}, acc3 = {};

    const int KT = K >> 5;
    issue_tile(0, 0);

    for (int kt = 0; kt < KT; ++kt) {
        const int buf = kt & 1;
        if (kt + 1 < KT) {
            issue_tile(kt + 1, buf ^ 1);
            asm volatile("s_wait_asynccnt 0x6" ::: "memory");  // oldest 6 done
        } else {
            asm volatile("s_wait_asynccnt 0x0" ::: "memory");
        }
        __syncthreads();

        const float* __restrict__ sAb = &sA[buf][mloc * LDS_STRIDE + khalf];
        const float* __restrict__ w0  = &sW[buf][(frow +  0) * LDS_STRIDE + khalf];
        const float* __restrict__ w1  = &sW[buf][(frow + 16) * LDS_STRIDE + khalf];
        const float* __restrict__ w2  = &sW[buf][(frow + 32) * LDS_STRIDE + khalf];
        const float* __restrict__ w3  = &sW[buf][(frow + 48) * LDS_STRIDE + khalf];

#pragma unroll
        for (int kk = 0; kk < 8; ++kk) {
            v2f a  = *(const v2f*)(sAb + kk * 4);
            v2f b0 = *(const v2f*)(w0 + kk * 4);
            v2f b1 = *(const v2f*)(w1 + kk * 4);
            v2f b2 = *(const v2f*)(w2 + kk * 4);
            v2f b3 = *(const v2f*)(w3 + kk * 4);
            acc0 = __builtin_amdgcn_wmma_f32_16x16x4_f32(false, a, false, b0, (short)0, acc0, false, false);
            acc1 = __builtin_amdgcn_wmma_f32_16x16x4_f32(false, a, false, b1, (short)0, acc1, false, false);
            acc2 = __builtin_amdgcn_wmma_f32_16x16x4_f32(false, a, false, b2, (short)0, acc2, false, false);
            acc3 = __builtin_amdgcn_wmma_f32_16x16x4_f32(false, a, false, b3, (short)0, acc3, false, false);
        }
        __syncthreads();
    }

    // --- store C (f32 C/D layout: v[i]: lanes 0-15 M=i, lanes 16-31 M=8+i) ---
    const int crow = row0 + wave * 16 + ((lane >> 4) << 3);
    const int ccol = col0 + frow;
    float* __restrict__ Cb = C + (size_t)crow * N + ccol;
#pragma unroll
    for (int i = 0; i < 8; ++i) {
        Cb[(size_t)i * N +  0] = acc0[i];
        Cb[(size_t)i * N + 16] = acc1[i];
        Cb[(size_t)i * N + 32] = acc2[i];
        Cb[(size_t)i * N + 48] = acc3[i];
    }
}

// ---------------------------------------------------------------------------
// Depthwise causal conv (K=4) + bias + SiLU over CONVD channels.
// ---------------------------------------------------------------------------
__global__ __launch_bounds__(256) void conv_silu_kernel(
    const float* __restrict__ proj, const float* __restrict__ conv_w,
    const float* __restrict__ conv_b, float* __restrict__ out)
{
    const int ch = blockIdx.x * 256 + threadIdx.x;   // 0..6143
    const int l  = blockIdx.y;                       // 0..511
    const float* __restrict__ w = conv_w + ch * KCONV;
    float acc = conv_b[ch];
#pragma unroll
    for (int j = 0; j < KCONV; ++j) {
        int ll = l - (KCONV - 1) + j;
        if (ll >= 0)
            acc += w[j] * proj[(size_t)ll * PROJ_ + INTER_ + ch];
    }
    out[(size_t)l * CONVD + ch] = acc / (1.f + __expf(-acc));   // SiLU
}

// ---------------------------------------------------------------------------
// dt = softplus(dt_raw + dt_bias); dA = exp(dt * (-exp(a_log)))
// ---------------------------------------------------------------------------
__global__ __launch_bounds__(256) void dt_kernel(
    const float* __restrict__ proj, const float* __restrict__ dt_bias,
    const float* __restrict__ a_log, float* __restrict__ dt,
    float* __restrict__ dA)
{
    const int idx = blockIdx.x * 256 + threadIdx.x;  // l*64 + h
    const int l = idx >> 6, h = idx & 63;
    float z  = proj[(size_t)l * PROJ_ + INTER_ + CONVD + h] + dt_bias[h];
    float sp = (z > 20.f) ? z : __logf(1.f + __expf(z));
    sp = fmaxf(sp, 0.f);
    float a  = -__expf(a_log[h]);
    dt[idx] = sp;
    dA[idx] = __expf(sp * a);
}

// ---------------------------------------------------------------------------
// Sequential selective scan. One block per head (64 blocks), 512 threads.
// thread t: p = t>>3, covers n = (t&7)*16 .. +16 (16 state regs).
// ---------------------------------------------------------------------------
__global__ __launch_bounds__(512) void scan_kernel(
    const float* __restrict__ conv, const float* __restrict__ dt,
    const float* __restrict__ dA, const float* __restrict__ dvec,
    float* __restrict__ y)
{
    const int h    = blockIdx.x;
    const int tid  = threadIdx.x;
    const int p    = tid >> 3;
    const int nsub = tid & 7;
    const int n0   = nsub * 16;
    const int g    = h >> 3;

    __shared__ float sb[N_];
    __shared__ float sc[N_];

    float st[16];
#pragma unroll
    for (int i = 0; i < 16; ++i) st[i] = 0.f;
    const float dh = dvec[h];

    for (int l = 0; l < L_; ++l) {
        const float* __restrict__ row = conv + (size_t)l * CONVD;
        if (tid < N_)            sb[tid]      = row[INTER_ + g * N_ + tid];
        else if (tid < 2 * N_)   sc[tid - N_] = row[INTER_ + GN_ + g * N_ + (tid - N_)];
        __syncthreads();

        const float xp  = row[h * P_ + p];
        const float dtl = dt[l * H_ + h];
        const float dAl = dA[l * H_ + h];

        float partial = 0.f;
#pragma unroll
        for (int i = 0; i < 16; ++i) {
            st[i] = st[i] * dAl + dtl * sb[n0 + i] * xp;
            partial += st[i] * sc[n0 + i];
        }
        partial += __shfl_xor(partial, 1, 32);
        partial += __shfl_xor(partial, 2, 32);
        partial += __shfl_xor(partial, 4, 32);
        if (nsub == 0)
            y[(size_t)l * INTER_ + h * P_ + p] = partial + dh * xp;
        __syncthreads();
    }
}

// ---------------------------------------------------------------------------
// y *= silu(gate); grouped RMS norm over GS=512; *= norm_w.
// ---------------------------------------------------------------------------
__global__ __launch_bounds__(256) void gatenorm_kernel(
    const float* __restrict__ ys, const float* __restrict__ proj,
    const float* __restrict__ norm_w, float* __restrict__ yn)
{
    const int g   = blockIdx.x;
    const int l   = blockIdx.y;
    const int tid = threadIdx.x;
    __shared__ float red[256];

    float v[2];
    int   idx[2];
    float ss = 0.f;
#pragma unroll
    for (int j = 0; j < 2; ++j) {
        int i = g * GS_ + tid + j * 256;
        idx[j] = i;
        float gate = proj[(size_t)l * PROJ_ + i];
        float sg   = gate / (1.f + __expf(-gate));
        float val  = ys[(size_t)l * INTER_ + i] * sg;
        v[j] = val;
        ss += val * val;
    }
    red[tid] = ss;
    __syncthreads();
    for (int off = 128; off > 0; off >>= 1) {
        if (tid < off) red[tid] += red[tid + off];
        __syncthreads();
    }
    const float rs = rsqrtf(red[0] * (1.f / GS_) + EPS_);
#pragma unroll
    for (int j = 0; j < 2; ++j)
        yn[(size_t)l * INTER_ + idx[j]] = v[j] * rs * norm_w[idx[j]];
}

// ---------------------------------------------------------------------------
// Host launcher
// ---------------------------------------------------------------------------
extern "C" void kernel_launch(void* const* d_in, const int* in_sizes, int n_in,
                              void* d_out, int out_size, void* d_ws, size_t ws_size,
                              hipStream_t stream)
{
    const float* x       = (const float*)d_in[0];
    const float* W_in    = (const float*)d_in[1];
    const float* W_out   = (const float*)d_in[2];
    const float* a_log   = (const float*)d_in[3];
    const float* dvec    = (const float*)d_in[4];
    const float* dt_bias = (const float*)d_in[5];
    const float* conv_w  = (const float*)d_in[6];
    const float* conv_b  = (const float*)d_in[7];
    const float* norm_w  = (const float*)d_in[8];
    float* out = (float*)d_out;

    float* ws = (float*)d_ws;
    float* proj  = ws;                                   // 512*10304
    float* conv  = proj + (size_t)L_ * PROJ_;            // 512*6144
    float* dt    = conv + (size_t)L_ * CONVD;            // 512*64
    float* dA    = dt   + (size_t)L_ * H_;               // 512*64
    float* ys    = dA   + (size_t)L_ * H_;               // 512*4096
    float* yn    = ys   + (size_t)L_ * INTER_;           // 512*4096

    // 1) proj = x @ W_in^T   (M=512, N=10304, K=4096)
    wmma_gemm_xWt<<<dim3(PROJ_ / 64, L_ / 128), 256, 0, stream>>>(
        x, W_in, proj, L_, PROJ_, HID_);

    // 2) depthwise conv + SiLU
    conv_silu_kernel<<<dim3(CONVD / 256, L_), 256, 0, stream>>>(
        proj, conv_w, conv_b, conv);

    // 3) dt / dA
    dt_kernel<<<(L_ * H_) / 256, 256, 0, stream>>>(proj, dt_bias, a_log, dt, dA);

    // 4) selective scan (one block per head)
    scan_kernel<<<H_, 512, 0, stream>>>(conv, dt, dA, dvec, ys);

    // 5) gate * silu + grouped RMS norm
    gatenorm_kernel<<<dim3(G_, L_), 256, 0, stream>>>(ys, proj, norm_w, yn);

    // 6) out = yn @ W_out^T  (M=512, N=4096, K=4096)
    wmma_gemm_xWt<<<dim3(HID_ / 64, L_ / 128), 256, 0, stream>>>(
        yn, W_out, out, L_, HID_, INTER_);
}